// DigitConvolutionalModel_35562329211261
// MI455X (gfx1250) — compile-verified
//
#include <hip/hip_runtime.h>
#include <stdint.h>

typedef __attribute__((ext_vector_type(2))) float v2f;
typedef __attribute__((ext_vector_type(8))) float v8f;

#define MT      64            // batch rows per workgroup
#define KCH     52            // K-chunk = 2 conv output rows (2*26)
#define NCHUNK  13            // 13*52 = 676
#define W1S     272           // LDS stride for W1 chunk   [52][272]
#define FS      53            // LDS stride for feat chunk [64][53]
#define HS      260           // LDS stride for h          [64][260]

// LDS layout (bytes)
#define OFF_W1B0  0                              // [52][272] f32 = 56576
#define OFF_W1B1  (OFF_W1B0 + KCH * W1S * 4)     // 56576
#define OFF_FEAT  (OFF_W1B1 + KCH * W1S * 4)     // 113152
#define OFF_W2P   (OFF_FEAT + MT * FS * 4)       // 126720 : [256][16] f32 = 16384
#define OFF_LOG   (OFF_W2P + 256 * 16 * 4)       // 143104 : [64][16] f32 = 4096
#define OFF_B1    (OFF_LOG + MT * 16 * 4)        // 147200
#define OFF_B2    (OFF_B1 + 256 * 4)             // 148224
#define SMEM_BYTES (OFF_B2 + 64)                 // 148288 B  (sH[64][260] aliases W1 bufs)

template<int W0>
__device__ __forceinline__ void conv_row13(const float (&r0)[16], const float (&r1)[16],
                                           const float (&r2)[16], const float* cw,
                                           float* frow) {
#pragma unroll
    for (int t = 0; t < 13; ++t) {
        frow[t] = r0[t + W0]     * cw[0] + r0[t + W0 + 1] * cw[1] + r0[t + W0 + 2] * cw[2]
                + r1[t + W0]     * cw[3] + r1[t + W0 + 1] * cw[4] + r1[t + W0 + 2] * cw[5]
                + r2[t + W0]     * cw[6] + r2[t + W0 + 1] * cw[7] + r2[t + W0 + 2] * cw[8];
    }
}

__global__ __launch_bounds__(256)
void digitconv_fused_kernel(const float* __restrict__ x,
                            const float* __restrict__ convw,
                            const float* __restrict__ W1g,
                            const float* __restrict__ b1g,
                            const float* __restrict__ W2g,
                            const float* __restrict__ b2g,
                            float* __restrict__ out,
                            int Btot)
{
    __shared__ __align__(16) unsigned char smem[SMEM_BYTES];
    float* sW1b0 = (float*)(smem + OFF_W1B0);
    float* sW1b1 = (float*)(smem + OFF_W1B1);
    float* sFeat = (float*)(smem + OFF_FEAT);
    float* sH    = (float*)(smem);              // aliases W1 buffers (epilogue only)
    float* sW2p  = (float*)(smem + OFF_W2P);    // [256][16], cols 10..15 zero
    float* sLog  = (float*)(smem + OFF_LOG);    // [64][16] partial logits
    float* sB1   = (float*)(smem + OFF_B1);
    float* sB2   = (float*)(smem + OFF_B2);

    const int tid   = threadIdx.x;
    const int lane  = tid & 31;
    const int wv    = tid >> 5;        // 0..7
    const int l15   = lane & 15;
    const int lh    = lane >> 4;       // half-wave (0/1)
    const int nBase = wv * 32;         // GEMM1: this wave's 32 N columns
    const int row0  = blockIdx.x * MT;

    // 3x3 conv weights (uniform)
    float cw[9];
#pragma unroll
    for (int i = 0; i < 9; ++i) cw[i] = convw[i];

    // One-time staging: padded W2 [256][16], b1, b2, zero sLog
    for (int i = tid; i < 4096; i += 256) {
        int n = i >> 4, c = i & 15;
        sW2p[i] = (c < 10) ? W2g[n * 10 + c] : 0.f;
    }
    sB1[tid] = b1g[tid];
    if (tid < 10) sB2[tid] = b2g[tid];
    for (int i = tid; i < MT * 16; i += 256) sLog[i] = 0.f;

    // ---- async stage of one W1 chunk [52][256] -> LDS [52][272] -------------
    auto issueW1 = [&](int jj, float* buf) {
        const float* src = W1g + (size_t)jj * KCH * 256;
        uint32_t ldsbase = (uint32_t)(uintptr_t)buf;
        for (int i = tid; i < KCH * 64; i += 256) {     // 64 float4 per row
            int kk = i >> 6, c4 = i & 63;
            uint64_t ga = (uint64_t)(uintptr_t)(src + kk * 256 + c4 * 4);
            uint32_t la = ldsbase + (uint32_t)((kk * W1S + c4 * 4) * 4);
            asm volatile("global_load_async_to_lds_b128 %0, %1, off"
                         :: "v"(la), "v"(ga) : "memory");
        }
    };

    const v8f vzero = {0.f, 0.f, 0.f, 0.f, 0.f, 0.f, 0.f, 0.f};
    v8f acc[4][2];
#pragma unroll
    for (int mt = 0; mt < 4; ++mt)
#pragma unroll
        for (int nt = 0; nt < 2; ++nt) acc[mt][nt] = vzero;

    issueW1(0, sW1b0);                 // prologue: chunk 0 in flight

    for (int j = 0; j < NCHUNK; ++j) {
        float* cur = (j & 1) ? sW1b1 : sW1b0;
        float* nxt = (j & 1) ? sW1b0 : sW1b1;

        __syncthreads();   // (a) WMMA(j-1) finished reading sFeat and `nxt`

        // ---- conv producer: 64 rows x 52 feats; thread = (m, dy, half) ----
        {
            int m    = tid >> 2;
            int sub  = tid & 3;
            int dy   = sub >> 1;
            int half = sub & 1;
            int oy   = 2 * j + dy;
            int c0   = half ? 12 : 0;      // loaded cols c0..c0+15 (16B aligned)
            int ox0  = half ? 13 : 0;      // outputs ox0..ox0+12

            float r0[16], r1[16], r2[16];
            if (row0 + m < Btot) {
                const float* xb = x + (size_t)(row0 + m) * 784 + oy * 28 + c0;
#pragma unroll
                for (int q = 0; q < 4; ++q) {
                    float4 a0 = *(const float4*)(xb + q * 4);
                    float4 a1 = *(const float4*)(xb + 28 + q * 4);
                    float4 a2 = *(const float4*)(xb + 56 + q * 4);
                    r0[q*4+0]=a0.x; r0[q*4+1]=a0.y; r0[q*4+2]=a0.z; r0[q*4+3]=a0.w;
                    r1[q*4+0]=a1.x; r1[q*4+1]=a1.y; r1[q*4+2]=a1.z; r1[q*4+3]=a1.w;
                    r2[q*4+0]=a2.x; r2[q*4+1]=a2.y; r2[q*4+2]=a2.z; r2[q*4+3]=a2.w;
                }
                if (j + 1 < NCHUNK) {      // warm next chunk's new x rows
                    __builtin_prefetch(xb + 84, 0, 0);
                    __builtin_prefetch(xb + 112, 0, 0);
                }
            } else {
#pragma unroll
                for (int q = 0; q < 16; ++q) { r0[q] = 0.f; r1[q] = 0.f; r2[q] = 0.f; }
            }
            float* frow = &sFeat[m * FS + dy * 26 + ox0];
            if (half) conv_row13<1>(r0, r1, r2, cw, frow);
            else      conv_row13<0>(r0, r1, r2, cw, frow);
        }

        // chunk j's async W1 copy (issued last iteration) must be in LDS
        asm volatile("s_wait_asynccnt 0x0" ::: "memory");
        __syncthreads();   // (b) publish sFeat(j) and cur W1 buffer

        if (j + 1 < NCHUNK) issueW1(j + 1, nxt);   // overlaps the WMMA block below

        // ---- GEMM1: 13 k-steps of V_WMMA_F32_16X16X4_F32 ----
        for (int ks = 0; ks < KCH; ks += 4) {
            v2f a[4], b[2];
#pragma unroll
            for (int mt = 0; mt < 4; ++mt) {
                const float* p = &sFeat[(mt * 16 + l15) * FS + ks + lh * 2];
                a[mt] = (v2f){p[0], p[1]};
            }
#pragma unroll
            for (int nt = 0; nt < 2; ++nt) {
                const float* q = &cur[(ks + lh * 2) * W1S + nBase + nt * 16 + l15];
                b[nt] = (v2f){q[0], q[W1S]};
            }
#pragma unroll
            for (int mt = 0; mt < 4; ++mt)
#pragma unroll
                for (int nt = 0; nt < 2; ++nt)
                    acc[mt][nt] = __builtin_amdgcn_wmma_f32_16x16x4_f32(
                        false, a[mt], false, b[nt],
                        (short)0, acc[mt][nt], false, false);
        }
    }

    __syncthreads();   // all WMMA reads of the aliased region complete

    // ---- bias + relu, stage h[64][256] into LDS ----
    // C/D layout: VGPR i: lanes0-15 -> M=i, lanes16-31 -> M=i+8; N = lane&15
#pragma unroll
    for (int mt = 0; mt < 4; ++mt) {
#pragma unroll
        for (int nt = 0; nt < 2; ++nt) {
            int n = nBase + nt * 16 + l15;
            float bn = sB1[n];
            v8f A = acc[mt][nt];
#pragma unroll
            for (int i = 0; i < 8; ++i) {
                float v = fmaxf(A[i] + bn, 0.f);
                sH[(mt * 16 + lh * 8 + i) * HS + n] = v;
            }
        }
    }

    __syncthreads();

    // ---- GEMM2 via WMMA: wave w reduces K-slice [32w, 32w+32) of h @ W2p ----
    {
        const int kBase = wv * 32;
        v8f acc2[4] = {vzero, vzero, vzero, vzero};
        for (int ks = 0; ks < 32; ks += 4) {
            const float* q = &sW2p[(kBase + ks + lh * 2) * 16 + l15];
            v2f b2 = (v2f){q[0], q[16]};
#pragma unroll
            for (int mt = 0; mt < 4; ++mt) {
                const float* p = &sH[(mt * 16 + l15) * HS + kBase + ks + lh * 2];
                v2f a2 = (v2f){p[0], p[1]};
                acc2[mt] = __builtin_amdgcn_wmma_f32_16x16x4_f32(
                    false, a2, false, b2, (short)0, acc2[mt], false, false);
            }
        }
        // cross-wave K reduction via LDS float atomics (ds_add_f32)
#pragma unroll
        for (int mt = 0; mt < 4; ++mt)
#pragma unroll
            for (int i = 0; i < 8; ++i)
                atomicAdd(&sLog[(mt * 16 + lh * 8 + i) * 16 + l15], acc2[mt][i]);
    }

    __syncthreads();

    // ---- + b2, write logits ----
    for (int ot = tid; ot < MT * 10; ot += 256) {
        int m = ot / 10;
        int c = ot - m * 10;
        if (row0 + m < Btot)
            out[(size_t)(row0 + m) * 10 + c] = sLog[m * 16 + c] + sB2[c];
    }
}

extern "C" void kernel_launch(void* const* d_in, const int* in_sizes, int n_in,
                              void* d_out, int out_size, void* d_ws, size_t ws_size,
                              hipStream_t stream) {
    (void)n_in; (void)out_size; (void)d_ws; (void)ws_size;
    const float* x  = (const float*)d_in[0];
    const float* cw = (const float*)d_in[1];
    const float* W1 = (const float*)d_in[2];
    const float* b1 = (const float*)d_in[3];
    const float* W2 = (const float*)d_in[4];
    const float* b2 = (const float*)d_in[5];
    float* out = (float*)d_out;

    int B = in_sizes[0] / 784;               // 32768
    int nblocks = (B + MT - 1) / MT;         // 512 workgroups
    hipLaunchKernelGGL(digitconv_fused_kernel, dim3(nblocks), dim3(256), 0, stream,
                       x, cw, W1, b1, W2, b2, out, B);
}